// EdgeFeaturizer_47974784696344
// MI455X (gfx1250) — compile-verified
//
#include <hip/hip_runtime.h>
#include <stdint.h>

// ---------------------------------------------------------------------------
// EdgeFeaturizer for MI455X (gfx1250).
//   - per-row top-12-smallest selection over an 8192x8192 fp32 matrix
//   - memory-bound: 256 MiB stream @ 23.3 TB/s ~= 11 us floor
//   - one wave32 per row; row streamed through LDS with double-buffered
//     global_load_async_to_lds_b128 (ASYNCcnt), candidates filtered by a
//     conservative threshold into an LDS buffer, exact top-12 selected by a
//     wave-wide u64 min tournament; deterministic exact fallback otherwise.
//   - s_wait_dscnt 0 before each async issue: DScnt and ASYNCcnt are mutually
//     unordered, so retire in-flight LDS reads before the async engine may
//     overwrite the staging buffer (WAR on LDS).
// ---------------------------------------------------------------------------

#define N_ATOMS        8192
#define KNB            12
#define NBINS          50
#define CHUNK          1024                 // floats per stage buffer (4 KB)
#define NCHUNK         (N_ATOMS / CHUNK)    // 8
#define CAP            512                  // candidate buffer entries per wave
#define WAVES_PER_BLK  4
#define THRESH         0.02f                // E[count] ~= 164 for U[0,1) rows

// key = (float bits of masked distance) << 32 | column.
// For non-negative floats the bit pattern is order-preserving, so u64
// comparison == (distance, index) lexicographic order — identical tie
// handling to jax.lax.top_k on -masked.
__device__ __forceinline__ unsigned long long makeKey(float d, int col) {
    unsigned int bits = (d <= 8.0f) ? __float_as_uint(d) : 0x7F800000u; // mask -> +inf
    return ((unsigned long long)bits << 32) | (unsigned int)col;
}

__device__ __forceinline__ unsigned long long waveMinU64(unsigned long long v) {
#pragma unroll
    for (int off = 16; off > 0; off >>= 1) {
        unsigned long long o = __shfl_xor(v, off, 32);
        v = (o < v) ? o : v;
    }
    return v;
}

// Async-copy CHUNK floats (4096 B) global -> LDS: 8 x b128, 16 B per lane each.
__device__ __forceinline__ void asyncCopyChunk(const float* gsrc,
                                               uint32_t ldsByteBase, int lane) {
    unsigned long long g = (unsigned long long)(uintptr_t)gsrc
                           + (unsigned)(lane * 16);
    uint32_t l = ldsByteBase + (uint32_t)(lane * 16);
#pragma unroll
    for (int j = 0; j < (CHUNK * 4) / 512; ++j) {   // 8 instructions
        asm volatile("global_load_async_to_lds_b128 %0, %1, off"
                     :: "v"(l), "v"(g) : "memory");
        g += 512;
        l += 512;
    }
}

__global__ __launch_bounds__(WAVES_PER_BLK * 32)
void edge_featurizer_kernel(const float* __restrict__ dm,
                            float* __restrict__ ei,   // [N*K*2] as float
                            float* __restrict__ ef) { // [N*K*NBINS]
    __shared__ __align__(16) float stage[WAVES_PER_BLK][2][CHUNK];  // 32 KB
    __shared__ unsigned long long cand[WAVES_PER_BLK][CAP];         // 16 KB
    __shared__ unsigned long long wsel[WAVES_PER_BLK][KNB];
    __shared__ unsigned int      candCnt[WAVES_PER_BLK];

    const int lane = (int)(threadIdx.x & 31u);
    const int w    = (int)(threadIdx.x >> 5);
    const int row  = (int)blockIdx.x * WAVES_PER_BLK + w;

    const float* __restrict__ rowPtr = dm + (size_t)row * N_ATOMS;

    if (lane == 0) candCnt[w] = 0;   // same-wave DS ops stay in order

    // Low 32 bits of a flat pointer into the LDS aperture == LDS byte offset.
    const uint32_t stageBase0 = (uint32_t)(uintptr_t)&stage[w][0][0];
    const uint32_t stageBase1 = (uint32_t)(uintptr_t)&stage[w][1][0];

    // ---- streaming scan with double-buffered async global->LDS copies ----
    asm volatile("s_wait_dscnt 0x0" ::: "memory");
    asyncCopyChunk(rowPtr, stageBase0, lane);

    for (int c = 0; c < NCHUNK; ++c) {
        if (c + 1 < NCHUNK) {
            // retire pending LDS reads before async engine may write LDS (WAR)
            asm volatile("s_wait_dscnt 0x0" ::: "memory");
            asyncCopyChunk(rowPtr + (size_t)(c + 1) * CHUNK,
                           ((c + 1) & 1) ? stageBase1 : stageBase0, lane);
            // up to 16 outstanding; in-order completion => first 8 (chunk c) done
            asm volatile("s_wait_asynccnt 0x8" ::: "memory");
        } else {
            asm volatile("s_wait_asynccnt 0x0" ::: "memory");
        }

        const float4* sp = (const float4*)&stage[w][c & 1][0];
#pragma unroll
        for (int j = 0; j < CHUNK / 4 / 32; ++j) {          // 8 iterations
            float4 v   = sp[j * 32 + lane];
            int   col0 = c * CHUNK + (j * 32 + lane) * 4;

            float dc[4];
            dc[0] = v.x; dc[1] = v.y; dc[2] = v.z; dc[3] = v.w;
#pragma unroll
            for (int q = 0; q < 4; ++q) {
                float d = dc[q];
                if (d <= THRESH) {
                    unsigned int pos = atomicAdd(&candCnt[w], 1u);
                    if (pos < CAP) cand[w][pos] = makeKey(d, col0 + q);
                }
            }
        }
    }

    // ---- exact top-12 selection ----
    const unsigned int cnt = candCnt[w];
    if (cnt >= KNB && cnt <= CAP) {
        // tournament over <=512 LDS candidates (keys are unique)
        for (int r = 0; r < KNB; ++r) {
            unsigned long long lm = ~0ull;
            int mpos = -1;
            for (int i = lane; i < (int)cnt; i += 32) {
                unsigned long long k = cand[w][i];
                if (k < lm) { lm = k; mpos = i; }
            }
            unsigned long long win = waveMinU64(lm);
            if (lm == win && mpos >= 0) cand[w][mpos] = ~0ull; // retire winner
            if (lane == 0) wsel[w][r] = win;
        }
    } else {
        // deterministic exact fallback: 12 min-passes over the row
        unsigned long long prevW = 0ull;
        for (int r = 0; r < KNB; ++r) {
            unsigned long long lm = ~0ull;
            for (int i = lane; i < N_ATOMS; i += 32) {
                unsigned long long k = makeKey(rowPtr[i], i);
                if ((r == 0 || k > prevW) && k < lm) lm = k;
            }
            unsigned long long win = waveMinU64(lm);
            if (lane == 0) wsel[w][r] = win;
            prevW = win;
        }
    }

    // ---- outputs ----
    // edge_index: [row*K + j][2] = (row, col_j), ascending distance order
    float* eiRow = ei + (size_t)row * (KNB * 2);
    if (lane < KNB * 2) {
        int j = lane >> 1;
        unsigned int col = (unsigned int)(wsel[w][j] & 0xFFFFFFFFull);
        eiRow[lane] = (lane & 1) ? (float)col : (float)row;
    }

    // features: Gaussian RBF of the *raw* selected distances (reference uses
    // take_along_axis on the unmasked matrix), centers = linspace(0,1,50)
    float* efRow = ef + (size_t)row * (KNB * NBINS);
    for (int t = lane; t < KNB * NBINS; t += 32) {
        int j = t / NBINS;
        int b = t - j * NBINS;
        unsigned int col = (unsigned int)(wsel[w][j] & 0xFFFFFFFFull);
        float d = rowPtr[col];
        float z = (d - (float)b * (1.0f / 49.0f)) * 5.0f;   // /WIDTH = *5
        efRow[t] = __expf(-0.5f * z * z);
    }
}

extern "C" void kernel_launch(void* const* d_in, const int* in_sizes, int n_in,
                              void* d_out, int out_size, void* d_ws, size_t ws_size,
                              hipStream_t stream) {
    (void)in_sizes; (void)n_in; (void)out_size; (void)d_ws; (void)ws_size;

    const float* dm = (const float*)d_in[0];
    float* out = (float*)d_out;
    float* ei  = out;                                     // 8192*12*2 elements
    float* ef  = out + (size_t)N_ATOMS * KNB * 2;         // 8192*12*50 elements

    dim3 grid(N_ATOMS / WAVES_PER_BLK);   // 2048 blocks
    dim3 block(WAVES_PER_BLK * 32);       // 128 threads = 4 wave32
    hipLaunchKernelGGL(edge_featurizer_kernel, grid, block, 0, stream,
                       dm, ei, ef);
}